// ClebschGordanTensorProductDense_56401510531322
// MI455X (gfx1250) — compile-verified
//
#include <hip/hip_runtime.h>
#include <math.h>

// ============================================================================
// ClebschGordan TP (64x0e+64x1o+64x2e) x (1x0e+1x1o+1x2e) -> grouped Linears.
//
// MI455X roofline: math ~28.6 GFLOP (grouped Linears), HBM traffic ~1 GB
// (x1 38MB + TP panels 2x340MB + out 340MB) => bandwidth-bound at 23.3 TB/s.
// Keep f32 exactness and use V_WMMA_F32_16X16X4_F32 for the
// (E*(2l+1)) x M x M GEMMs; each group's X panel (<=84MB) fits in 192MB L2,
// so column-tile A re-reads stay on-chip.
//
// Phase 1 (tp_kernel): per-edge CG contraction. CG tables are evaluated at
// COMPILE TIME (constexpr port of the reference's Racah formula + real-SH
// basis transform), so coefficients fold to immediates and zeros vanish.
// Phase 2 (wmma_gemm<L3>): Y = (1/sqrt(M)) * X * W, one wave per 16x64 strip;
// the 64x64 block tile is staged in LDS and written back with fully
// coalesced stores ordered by the e3nn output layout (v*L3+m contiguous).
// ============================================================================

typedef __attribute__((ext_vector_type(2))) float v2f;
typedef __attribute__((ext_vector_type(8))) float v8f;

#define E_EDGES 16384
#define OUT_DIM 5184

// --------------------------- compile-time CG -------------------------------
namespace cgc {

constexpr double FT[11] = {1.0, 1.0, 2.0, 6.0, 24.0, 120.0, 720.0, 5040.0,
                           40320.0, 362880.0, 3628800.0};

constexpr double csqrt(double x) {
  if (x <= 0.0) return 0.0;
  double g = x > 1.0 ? x : 1.0;
  for (int i = 0; i < 48; ++i) g = 0.5 * (g + x / g);
  return g;
}

constexpr double su2cg(int j1, int m1, int j2, int m2, int j3, int m3) {
  if (m3 != m1 + m2) return 0.0;
  const int jmin = j1 > j2 ? j1 - j2 : j2 - j1;
  if (j3 < jmin || j3 > j1 + j2) return 0.0;
  double pre = csqrt((2.0 * j3 + 1.0) * FT[j1 + j2 - j3] * FT[j1 - j2 + j3] *
                     FT[-j1 + j2 + j3] / FT[j1 + j2 + j3 + 1]);
  pre *= csqrt(FT[j1 + m1] * FT[j1 - m1] * FT[j2 + m2] * FT[j2 - m2] *
               FT[j3 + m3] * FT[j3 - m3]);
  int kmin = 0;
  if (j2 - j3 - m1 > kmin) kmin = j2 - j3 - m1;
  if (j1 - j3 + m2 > kmin) kmin = j1 - j3 + m2;
  int kmax = j1 + j2 - j3;
  if (j1 - m1 < kmax) kmax = j1 - m1;
  if (j2 + m2 < kmax) kmax = j2 + m2;
  double s = 0.0;
  for (int k = kmin; k <= kmax; ++k) {
    const double d = FT[k] * FT[j1 + j2 - j3 - k] * FT[j1 - m1 - k] *
                     FT[j2 + m2 - k] * FT[j3 - j2 + m1 + k] * FT[j3 - j1 - m2 + k];
    s += ((k & 1) ? -1.0 : 1.0) / d;
  }
  return pre * s;
}

struct Cx { double re; double im; };
constexpr Cx cxmul(Cx a, Cx b) {
  return Cx{a.re * b.re - a.im * b.im, a.re * b.im + a.im * b.re};
}

struct URow { int n; int col[2]; Cx val[2]; };

// Nonzero entries of row i of the real<-complex SH change-of-basis U(l).
constexpr URow urow(int l, int i) {
  URow r{};
  const int m = i - l;
  const double s2 = csqrt(2.0);
  if (m < 0) {
    const double sg = ((-m) & 1) ? -1.0 : 1.0;  // (-1)^m
    r.n = 2;
    r.col[0] = l + m; r.val[0] = Cx{0.0, 1.0 / s2};
    r.col[1] = l - m; r.val[1] = Cx{0.0, -sg / s2};
  } else if (m == 0) {
    r.n = 1; r.col[0] = l; r.val[0] = Cx{1.0, 0.0};
  } else {
    const double sg = (m & 1) ? -1.0 : 1.0;
    r.n = 2;
    r.col[0] = l - m; r.val[0] = Cx{1.0 / s2, 0.0};
    r.col[1] = l + m; r.val[1] = Cx{sg / s2, 0.0};
  }
  return r;
}

template <int L1, int L2, int L3>
struct CGTab { float v[(2 * L1 + 1) * (2 * L2 + 1) * (2 * L3 + 1)]; };

template <int L1, int L2, int L3>
constexpr CGTab<L1, L2, L3> build() {
  constexpr int D1 = 2 * L1 + 1, D2 = 2 * L2 + 1, D3 = 2 * L3 + 1;
  URow u1[D1] = {}; URow u2[D2] = {}; URow u3[D3] = {};
  for (int i = 0; i < D1; ++i) u1[i] = urow(L1, i);
  for (int j = 0; j < D2; ++j) u2[j] = urow(L2, j);
  for (int k = 0; k < D3; ++k) u3[k] = urow(L3, k);
  double tre[D1 * D2 * D3] = {};
  double tim[D1 * D2 * D3] = {};
  for (int i = 0; i < D1; ++i)
    for (int j = 0; j < D2; ++j)
      for (int k = 0; k < D3; ++k) {
        Cx s{0.0, 0.0};
        for (int a = 0; a < u1[i].n; ++a)
          for (int b = 0; b < u2[j].n; ++b)
            for (int c = 0; c < u3[k].n; ++c) {
              const int m1 = u1[i].col[a] - L1;
              const int m2 = u2[j].col[b] - L2;
              const int m3 = u3[k].col[c] - L3;
              if (m3 != m1 + m2) continue;
              const double cg = su2cg(L1, m1, L2, m2, L3, m3);
              if (cg == 0.0) continue;
              Cx t = cxmul(u1[i].val[a], u2[j].val[b]);
              t = cxmul(t, Cx{u3[k].val[c].re, -u3[k].val[c].im});  // conj(U3)
              s.re += t.re * cg; s.im += t.im * cg;
            }
        tre[(i * D2 + j) * D3 + k] = s.re;
        tim[(i * D2 + j) * D3 + k] = s.im;
      }
  double nre = 0.0, nim = 0.0;
  for (int t = 0; t < D1 * D2 * D3; ++t) { nre += tre[t] * tre[t]; nim += tim[t] * tim[t]; }
  const bool useRe = nre >= nim;
  const double sc = csqrt(2.0 * L3 + 1.0) / csqrt(useRe ? nre : nim);
  CGTab<L1, L2, L3> out{};
  for (int t = 0; t < D1 * D2 * D3; ++t)
    out.v[t] = (float)((useRe ? tre[t] : tim[t]) * sc);
  return out;
}

}  // namespace cgc

static constexpr auto CG_000 = cgc::build<0, 0, 0>();
static constexpr auto CG_011 = cgc::build<0, 1, 1>();
static constexpr auto CG_022 = cgc::build<0, 2, 2>();
static constexpr auto CG_101 = cgc::build<1, 0, 1>();
static constexpr auto CG_110 = cgc::build<1, 1, 0>();
static constexpr auto CG_111 = cgc::build<1, 1, 1>();
static constexpr auto CG_112 = cgc::build<1, 1, 2>();
static constexpr auto CG_121 = cgc::build<1, 2, 1>();
static constexpr auto CG_122 = cgc::build<1, 2, 2>();
static constexpr auto CG_123 = cgc::build<1, 2, 3>();
static constexpr auto CG_202 = cgc::build<2, 0, 2>();
static constexpr auto CG_211 = cgc::build<2, 1, 1>();
static constexpr auto CG_212 = cgc::build<2, 1, 2>();
static constexpr auto CG_213 = cgc::build<2, 1, 3>();
static constexpr auto CG_220 = cgc::build<2, 2, 0>();
static constexpr auto CG_221 = cgc::build<2, 2, 1>();
static constexpr auto CG_222 = cgc::build<2, 2, 2>();
static constexpr auto CG_223 = cgc::build<2, 2, 3>();
static constexpr auto CG_224 = cgc::build<2, 2, 4>();

// Workspace panel offsets (floats); panel g is row-major (E*L3[g], M[g]).
#define XOFF0 0ull
#define XOFF1 3145728ull
#define XOFF2 15728640ull
#define XOFF3 22020096ull
#define XOFF4 32505856ull
#define XOFF5 53477376ull
#define XOFF6 68157440ull
#define XOFF7 75497472ull

// ------------------------------ phase 1: TP --------------------------------

template <int D1, int D2, int D3, int N>
__device__ __forceinline__ void cg_path(const float (&A)[D1], const float (&B)[D2],
                                        const float (&cg)[N],
                                        float* __restrict__ dst, size_t rowBase,
                                        int Mg, int col) {
  static_assert(N == D1 * D2 * D3, "size mismatch");
#pragma unroll
  for (int k = 0; k < D3; ++k) {
    float acc = 0.f;
#pragma unroll
    for (int i = 0; i < D1; ++i) {
#pragma unroll
      for (int j = 0; j < D2; ++j) {
        const float c = cg[(i * D2 + j) * D3 + k];
        if (c != 0.f) acc = fmaf(A[i] * B[j], c, acc);
      }
    }
    dst[(rowBase + (size_t)k) * (size_t)Mg + (size_t)col] = acc;
  }
}

__global__ __launch_bounds__(256) void tp_kernel(const float* __restrict__ x1,
                                                 const float* __restrict__ x2,
                                                 float* __restrict__ X) {
  const int tid = threadIdx.x;
  const int u = tid & 63;        // mul index 0..63
  const int el = tid >> 6;       // edge-in-block 0..3
  const size_t e = (size_t)blockIdx.x * 4 + el;

  __shared__ float sx2[4][12];
  if (u < 9) sx2[el][u] = x2[e * 9 + u];
  __syncthreads();

  const float* __restrict__ x1e = x1 + e * 576;
  float A0[1] = {x1e[u]};
  float A1[3];
#pragma unroll
  for (int i = 0; i < 3; ++i) A1[i] = x1e[64 + u * 3 + i];
  float A2[5];
#pragma unroll
  for (int i = 0; i < 5; ++i) A2[i] = x1e[256 + u * 5 + i];

  float B0[1] = {sx2[el][0]};
  float B1[3] = {sx2[el][1], sx2[el][2], sx2[el][3]};
  float B2[5] = {sx2[el][4], sx2[el][5], sx2[el][6], sx2[el][7], sx2[el][8]};

  // group meta: g0(M192,L1) g1(M256,L3) g2(M128,L3) g3(M128,L5)
  //             g4(M256,L5) g5(M128,L7) g6(M64,L7)  g7(M64,L9)
  cg_path<1, 1, 1>(A0, B0, CG_000.v, X + XOFF0, e * 1, 192, u);        // g0 s0
  cg_path<1, 3, 3>(A0, B1, CG_011.v, X + XOFF1, e * 3, 256, u);        // g1 s0
  cg_path<1, 5, 5>(A0, B2, CG_022.v, X + XOFF4, e * 5, 256, u);        // g4 s0
  cg_path<3, 1, 3>(A1, B0, CG_101.v, X + XOFF1, e * 3, 256, 64 + u);   // g1 s1
  cg_path<3, 3, 1>(A1, B1, CG_110.v, X + XOFF0, e * 1, 192, 64 + u);   // g0 s1
  cg_path<3, 3, 3>(A1, B1, CG_111.v, X + XOFF2, e * 3, 128, u);        // g2 s0
  cg_path<3, 3, 5>(A1, B1, CG_112.v, X + XOFF4, e * 5, 256, 64 + u);   // g4 s1
  cg_path<3, 5, 3>(A1, B2, CG_121.v, X + XOFF1, e * 3, 256, 128 + u);  // g1 s2
  cg_path<3, 5, 5>(A1, B2, CG_122.v, X + XOFF3, e * 5, 128, u);        // g3 s0
  cg_path<3, 5, 7>(A1, B2, CG_123.v, X + XOFF5, e * 7, 128, u);        // g5 s0
  cg_path<5, 1, 5>(A2, B0, CG_202.v, X + XOFF4, e * 5, 256, 128 + u);  // g4 s2
  cg_path<5, 3, 3>(A2, B1, CG_211.v, X + XOFF1, e * 3, 256, 192 + u);  // g1 s3
  cg_path<5, 3, 5>(A2, B1, CG_212.v, X + XOFF3, e * 5, 128, 64 + u);   // g3 s1
  cg_path<5, 3, 7>(A2, B1, CG_213.v, X + XOFF5, e * 7, 128, 64 + u);   // g5 s1
  cg_path<5, 5, 1>(A2, B2, CG_220.v, X + XOFF0, e * 1, 192, 128 + u);  // g0 s2
  cg_path<5, 5, 3>(A2, B2, CG_221.v, X + XOFF2, e * 3, 128, 64 + u);   // g2 s1
  cg_path<5, 5, 5>(A2, B2, CG_222.v, X + XOFF4, e * 5, 256, 192 + u);  // g4 s3
  cg_path<5, 5, 7>(A2, B2, CG_223.v, X + XOFF6, e * 7, 64, u);         // g6 s0
  cg_path<5, 5, 9>(A2, B2, CG_224.v, X + XOFF7, e * 9, 64, u);         // g7 s0
}

// --------------------------- phase 2: WMMA GEMM ----------------------------
// Y[r, v] = scale * sum_u X[r, u] * W[u, v];  r = e*L3 + m.
// Block = 4 waves = 64x64 output tile (each wave a 16x64 strip, 4 accums,
// A-frag reused 4x). Tile is staged in LDS, then written back coalesced:
// for each edge e inside the tile, the out range {v*L3+m} is one contiguous
// 64*L3-float segment.

template <int L3>
__global__ __launch_bounds__(128) void wmma_gemm(const float* __restrict__ X,
                                                 const float* __restrict__ W,
                                                 float* __restrict__ out,
                                                 int M, int outColOff, float scale) {
  const int tid = threadIdx.x;
  const int lane = tid & 31;
  const int wv = tid >> 5;
  const int hl = lane >> 4;   // half-wave: selects K pair (A/B) / +8 rows (D)
  const int lm = lane & 15;

  const int rblk = (int)blockIdx.x * 64;          // block row base
  const size_t r0 = (size_t)rblk + (size_t)wv * 16;  // wave row base
  const int n0 = (int)blockIdx.y * 64;

  __shared__ float tile[64][65];  // 65-float pitch -> conflict-free

  v8f acc0 = {0.f, 0.f, 0.f, 0.f, 0.f, 0.f, 0.f, 0.f};
  v8f acc1 = acc0, acc2 = acc0, acc3 = acc0;

  const float* __restrict__ arow = X + (r0 + (size_t)lm) * (size_t)M;
  const float* __restrict__ wb = W + n0 + lm;

  for (int k0 = 0; k0 < M; k0 += 4) {
    const int ka = k0 + 2 * hl;
    v2f a;
    a.x = arow[ka];
    a.y = arow[ka + 1];
    const float* w0 = wb + (size_t)ka * M;
    const float* w1 = w0 + M;
    v2f b0 = {w0[0], w1[0]};
    v2f b1 = {w0[16], w1[16]};
    v2f b2 = {w0[32], w1[32]};
    v2f b3 = {w0[48], w1[48]};
    acc0 = __builtin_amdgcn_wmma_f32_16x16x4_f32(false, a, false, b0, (short)0, acc0, false, false);
    acc1 = __builtin_amdgcn_wmma_f32_16x16x4_f32(false, a, false, b1, (short)0, acc1, false, false);
    acc2 = __builtin_amdgcn_wmma_f32_16x16x4_f32(false, a, false, b2, (short)0, acc2, false, false);
    acc3 = __builtin_amdgcn_wmma_f32_16x16x4_f32(false, a, false, b3, (short)0, acc3, false, false);
  }

  // Stage the wave's 16x64 strip into the block tile.
#pragma unroll
  for (int i = 0; i < 8; ++i) {
    const int lr = wv * 16 + i + 8 * hl;  // D row: VGPR i, +8 for hi half
    tile[lr][lm] = acc0[i];
    tile[lr][16 + lm] = acc1[i];
    tile[lr][32 + lm] = acc2[i];
    tile[lr][48 + lm] = acc3[i];
  }
  __syncthreads();

  // Coalesced write-back in e3nn output order.
  const int eFirst = rblk / L3;
  const int eLast = (rblk + 63) / L3;
  for (int e = eFirst; e <= eLast; ++e) {
    int mlo = rblk - e * L3;           // >0 only possible for eFirst
    if (mlo < 0) mlo = 0;
    int mhi = rblk + 64 - e * L3;      // <L3 only possible for eLast
    if (mhi > L3) mhi = L3;
    const int w = mhi - mlo;
    const int lrbase = e * L3 - rblk;
    float* __restrict__ outp =
        out + (size_t)e * OUT_DIM + (size_t)outColOff + (size_t)n0 * L3;
    if (w == L3) {
      // Interior edge: out offsets v*L3+mm == off, fully contiguous.
      for (int off = tid; off < 64 * L3; off += 128) {
        const int v = off / L3;
        const int mm = off - v * L3;
        outp[off] = tile[lrbase + mm][v] * scale;
      }
    } else {
      // Ragged tile-boundary edge (rare): contiguous per-v runs of length w.
      const int cnt = w * 64;
      for (int off = tid; off < cnt; off += 128) {
        const int v = off / w;
        const int mm = mlo + (off - v * w);
        outp[(size_t)v * L3 + mm] = tile[lrbase + mm][v] * scale;
      }
    }
  }
}

// ------------------------------- launcher ----------------------------------

extern "C" void kernel_launch(void* const* d_in, const int* in_sizes, int n_in,
                              void* d_out, int out_size, void* d_ws, size_t ws_size,
                              hipStream_t stream) {
  (void)in_sizes; (void)n_in; (void)out_size; (void)ws_size;
  const float* x1 = (const float*)d_in[0];
  const float* x2 = (const float*)d_in[1];
  float* X = (float*)d_ws;   // 84,934,656 floats = ~340 MB of panels
  float* out = (float*)d_out;

  tp_kernel<<<E_EDGES / 4, 256, 0, stream>>>(x1, x2, X);

  struct GInfo { size_t xoff; int R; int M; int L3; int coff; };
  static const GInfo G[8] = {
      {XOFF0, 16384, 192, 1, 0},     // (0,e): W_0e
      {XOFF1, 49152, 256, 3, 192},   // (1,o): W_1o
      {XOFF2, 49152, 128, 3, 960},   // (1,e): W_1e
      {XOFF3, 81920, 128, 5, 1344},  // (2,o): W_2o
      {XOFF4, 81920, 256, 5, 1984},  // (2,e): W_2e
      {XOFF5, 114688, 128, 7, 3264}, // (3,o): W_3o
      {XOFF6, 114688, 64, 7, 4160},  // (3,e): W_3e
      {XOFF7, 147456, 64, 9, 4608},  // (4,e): W_4e
  };

  for (int g = 0; g < 8; ++g) {
    const float* W = (const float*)d_in[2 + g];
    const float* Xg = X + G[g].xoff;
    const float scale = 1.0f / sqrtf((float)G[g].M);
    dim3 grid((unsigned)(G[g].R / 64), (unsigned)(G[g].M / 64));
    switch (G[g].L3) {
      case 1: wmma_gemm<1><<<grid, 128, 0, stream>>>(Xg, W, out, G[g].M, G[g].coff, scale); break;
      case 3: wmma_gemm<3><<<grid, 128, 0, stream>>>(Xg, W, out, G[g].M, G[g].coff, scale); break;
      case 5: wmma_gemm<5><<<grid, 128, 0, stream>>>(Xg, W, out, G[g].M, G[g].coff, scale); break;
      case 7: wmma_gemm<7><<<grid, 128, 0, stream>>>(Xg, W, out, G[g].M, G[g].coff, scale); break;
      case 9: wmma_gemm<9><<<grid, 128, 0, stream>>>(Xg, W, out, G[g].M, G[g].coff, scale); break;
    }
  }
}